// MSSIMLoss_18691697672269
// MI455X (gfx1250) — compile-verified
//
#include <hip/hip_runtime.h>

typedef __attribute__((ext_vector_type(2))) float v2f;
typedef __attribute__((ext_vector_type(8))) float v8f;

#define NB 8
#define DD 32
#define HH 256
#define WW 256
#define TOTAL_ELEMS (NB * DD * HH * WW)   // 16777216

#define ZS  34   // padded LDS stride (even => float2-aligned, conflict-free)
#define T1S 34

// ---------------------------------------------------------------------------
// Fused 3D-SSIM tile kernel.
// One block (160 threads = 5 wave32) -> one 16x16 (y,x) output tile at (n,z).
// Wave f owns field f in {img1, img2, img1^2, img2^2, img1*img2}:
//   z-pass : 5-tap VALU conv into LDS (stored column-major: [x][y])
//   y-pass : band(16x32) x zconv(32x32) as 16 v_wmma_f32_16x16x4_f32
//   x-pass : t1(16x32) x band(32x16)   as  8 v_wmma_f32_16x16x4_f32
// All WMMA fragments are contiguous 8B-aligned float2 in LDS -> ds_load_b64.
// ---------------------------------------------------------------------------
__global__ __launch_bounds__(160) void ssim3d_fused(
    const float* __restrict__ img1,
    const float* __restrict__ img2,
    float* __restrict__ acc)
{
  __shared__ float wz_s[5];
  __shared__ float wpad_s[48];         // wpad[16+d] = wy[d] for d in [0,10], else 0
  __shared__ float zc_s[5][32 * ZS];   // z-convolved fields, COLUMN-major [x][y]
  __shared__ float t1_s[5][16 * T1S];  // after y-pass: row-major [y][x halo]
  __shared__ float of_s[5][16][16];    // after x-pass: final conv outputs
  __shared__ float red_s[5];

  const int tid = threadIdx.x;
  const int x0 = blockIdx.x * 16;
  const int y0 = blockIdx.y * 16;
  const int bz = blockIdx.z;
  const int n  = bz >> 5;
  const int z  = bz & 31;

  // ---- speculative prefetch of this block's halo rows (1 row / thread) ---
  // 160 threads == 5 z-planes * 32 halo rows.
  {
    const int pl = tid >> 5;           // 0..4
    const int yl = tid & 31;           // 0..31
    const int gz = z - 2 + pl;
    const int gy = y0 - 5 + yl;
    const int gx = (x0 - 5) < 0 ? 0 : (x0 - 5);
    if ((unsigned)gz < (unsigned)DD && (unsigned)gy < (unsigned)HH) {
      const size_t ix =
          (((size_t)(n * DD + gz)) * HH + (size_t)gy) * WW + (size_t)gx;
      __builtin_prefetch(&img1[ix], 0, 0);
      __builtin_prefetch(&img2[ix], 0, 0);
    }
  }

  // ---- phase 0: Gaussian window weights (tiny, thread 0) -----------------
  if (tid == 0) {
    float s = 0.f;
    float wz[5];
    for (int i = 0; i < 5; ++i) {
      float d = (float)(i - 2);
      wz[i] = __expf(-d * d / (2.f * 0.7f * 0.7f));
      s += wz[i];
    }
    for (int i = 0; i < 5; ++i) wz_s[i] = wz[i] / s;
    s = 0.f;
    float wy[11];
    for (int i = 0; i < 11; ++i) {
      float d = (float)(i - 5);
      wy[i] = __expf(-d * d / (2.f * 1.5f * 1.5f));
      s += wy[i];
    }
    for (int i = 0; i < 48; ++i) wpad_s[i] = 0.f;
    for (int i = 0; i < 11; ++i) wpad_s[16 + i] = wy[i] / s;
  }
  __syncthreads();

  // ---- phase 1: load 32x32 halo, build 5 fields, 5-tap z-conv (VALU) -----
  {
    float wzr[5];
    #pragma unroll
    for (int i = 0; i < 5; ++i) wzr[i] = wz_s[i];

    for (int p = tid; p < 32 * 32; p += 160) {
      const int yl = p >> 5, xl = p & 31;   // xl fastest => coalesced global
      const int gy = y0 - 5 + yl;
      const int gx = x0 - 5 + xl;
      float a0 = 0.f, a1 = 0.f, a2 = 0.f, a3 = 0.f, a4 = 0.f;
      if ((unsigned)gy < (unsigned)HH && (unsigned)gx < (unsigned)WW) {
        #pragma unroll
        for (int dz = 0; dz < 5; ++dz) {
          const int gz = z - 2 + dz;
          if ((unsigned)gz < (unsigned)DD) {
            const size_t ix =
                (((size_t)(n * DD + gz)) * HH + (size_t)gy) * WW + (size_t)gx;
            const float a = img1[ix];
            const float b = img2[ix];
            const float w = wzr[dz];
            a0 = fmaf(w, a,     a0);
            a1 = fmaf(w, b,     a1);
            a2 = fmaf(w, a * a, a2);
            a3 = fmaf(w, b * b, a3);
            a4 = fmaf(w, a * b, a4);
          }
        }
      }
      // column-major store: y contiguous (stride-34 across lanes, no conflicts)
      zc_s[0][xl * ZS + yl] = a0;
      zc_s[1][xl * ZS + yl] = a1;
      zc_s[2][xl * ZS + yl] = a2;
      zc_s[3][xl * ZS + yl] = a3;
      zc_s[4][xl * ZS + yl] = a4;
    }
  }

  const int f    = tid >> 5;    // wave id == field id (5 waves)
  const int lane = tid & 31;
  const int half = lane >> 4;   // K-group select (lanes 16-31 hold K+2)
  const int lm   = lane & 15;   // M (A) / N (B) index

  // ---- hoist band fragments into registers (shared by y- and x-pass) -----
  // wb[kk] holds band(k0 - lm), band(k0+1 - lm) with k0 = 4*kk + 2*half.
  v2f wb[8];
  #pragma unroll
  for (int kk = 0; kk < 8; ++kk) {
    const int k0 = 4 * kk + 2 * half;
    wb[kk].x = wpad_s[16 + k0     - lm];
    wb[kk].y = wpad_s[16 + k0 + 1 - lm];
  }
  __syncthreads();   // zc_s ready

  // ---- phase 2: y-pass = band(16x32) x zconv(32x32) via WMMA -------------
  // B fragment rows (k0, k0+1) at fixed x-column are contiguous in zc_s.
  {
    v8f c0 = {};  // output x columns 0..15
    v8f c1 = {};  // output x columns 16..31
    #pragma unroll
    for (int kk = 0; kk < 8; ++kk) {
      const int k0 = 4 * kk + 2 * half;
      const v2f b0 = *(const v2f*)&zc_s[f][lm        * ZS + k0];
      const v2f b1 = *(const v2f*)&zc_s[f][(16 + lm) * ZS + k0];
      c0 = __builtin_amdgcn_wmma_f32_16x16x4_f32(
          false, wb[kk], false, b0, (short)0, c0, false, false);
      c1 = __builtin_amdgcn_wmma_f32_16x16x4_f32(
          false, wb[kk], false, b1, (short)0, c1, false, false);
    }
    #pragma unroll
    for (int r = 0; r < 8; ++r) {
      const int m = r + 8 * half;            // C/D layout: VGPR r, lane half
      t1_s[f][m * T1S + lm]      = c0[r];
      t1_s[f][m * T1S + 16 + lm] = c1[r];
    }
  }
  __syncthreads();

  // ---- phase 3: x-pass = t1(16x32) x band(32x16) via WMMA ----------------
  // A fragment (k0, k0+1) at fixed y-row is contiguous in t1_s.
  {
    v8f d = {};
    #pragma unroll
    for (int kk = 0; kk < 8; ++kk) {
      const int k0 = 4 * kk + 2 * half;
      const v2f a = *(const v2f*)&t1_s[f][lm * T1S + k0];
      d = __builtin_amdgcn_wmma_f32_16x16x4_f32(
          false, a, false, wb[kk], (short)0, d, false, false);
    }
    #pragma unroll
    for (int r = 0; r < 8; ++r)
      of_s[f][r + 8 * half][lm] = d[r];
  }
  __syncthreads();

  // ---- phase 4: SSIM map + block reduction -------------------------------
  {
    float local = 0.f;
    for (int p = tid; p < 256; p += 160) {
      const int m  = p >> 4;
      const int nx = p & 15;
      const float mu1 = of_s[0][m][nx];
      const float mu2 = of_s[1][m][nx];
      const float x11 = of_s[2][m][nx];
      const float x22 = of_s[3][m][nx];
      const float x12 = of_s[4][m][nx];
      const float mu1s = mu1 * mu1;
      const float mu2s = mu2 * mu2;
      const float mu12 = mu1 * mu2;
      const float s1  = x11 - mu1s;
      const float s2  = x22 - mu2s;
      const float s12 = x12 - mu12;
      const float C1 = 1e-4f, C2 = 9e-4f;
      local += ((2.f * mu12 + C1) * (2.f * s12 + C2)) /
               ((mu1s + mu2s + C1) * (s1 + s2 + C2));
    }
    #pragma unroll
    for (int off = 16; off > 0; off >>= 1)
      local += __shfl_xor(local, off, 32);
    if (lane == 0) red_s[f] = local;
  }
  __syncthreads();
  if (tid == 0) {
    float s = 0.f;
    #pragma unroll
    for (int i = 0; i < 5; ++i) s += red_s[i];
    atomicAdd(acc, s);
  }
}

__global__ void ssim3d_init(float* acc) { acc[0] = 0.f; }

__global__ void ssim3d_finalize(const float* __restrict__ acc,
                                float* __restrict__ out) {
  out[0] = 1.f - acc[0] * (1.f / (float)TOTAL_ELEMS);
}

extern "C" void kernel_launch(void* const* d_in, const int* in_sizes, int n_in,
                              void* d_out, int out_size, void* d_ws, size_t ws_size,
                              hipStream_t stream) {
  const float* img1 = (const float*)d_in[0];
  const float* img2 = (const float*)d_in[1];
  float* out = (float*)d_out;
  float* acc = (float*)d_ws;

  ssim3d_init<<<dim3(1), dim3(1), 0, stream>>>(acc);
  dim3 grid(WW / 16, HH / 16, NB * DD);   // 16 x 16 x 256 tiles
  dim3 block(160);                        // 5 wave32, one field per wave
  ssim3d_fused<<<grid, block, 0, stream>>>(img1, img2, acc);
  ssim3d_finalize<<<dim3(1), dim3(1), 0, stream>>>(acc, out);
}